// EncoderDecoder_66271345377340
// MI455X (gfx1250) — compile-verified
//
#include <hip/hip_runtime.h>
#include <hip/hip_bf16.h>

typedef __attribute__((ext_vector_type(2))) float v2f;
typedef __attribute__((ext_vector_type(8))) float v8f;
typedef __attribute__((ext_vector_type(4))) unsigned int v4u;
typedef __attribute__((ext_vector_type(4))) int v4i;
typedef __attribute__((ext_vector_type(8))) int v8i;

// Problem constants (from reference setup_inputs)
#define B_SRC 4096
#define S_SRC 32
#define DIM   512
#define B_TGT 32
#define TLEN  256
#define ROWS  (B_TGT * TLEN)   // 8192
#define COLS  B_SRC            // 4096
#define EPS   1e-12f

// LDS A-tile row stride in floats: 512 data + 2x2-dword TDM pads = 516.
// 516 mod 64 == 4 -> the 16 A-fragment lanes land on 16 distinct banks.
#define A_STRIDE 516

// ---------------------------------------------------------------------------
// Kernel 1: src_readout[b][d] = sum_s (mask/sum) * m[b][s][d];  also s_sq[b]
// Dominant HBM stream (268 MB of m) -> float2-vectorized loads.
// ---------------------------------------------------------------------------
__global__ __launch_bounds__(256) void k_src_readout(
    const float* __restrict__ m, const int* __restrict__ smask,
    float* __restrict__ s_read, float* __restrict__ s_sq) {
  __shared__ float w[S_SRC];
  __shared__ float red[256];
  const int b = blockIdx.x, tid = threadIdx.x;
  if (tid < S_SRC) w[tid] = (float)smask[b * S_SRC + tid];
  __syncthreads();
  float msum = 0.f;
#pragma unroll
  for (int s = 0; s < S_SRC; ++s) msum += w[s];
  const float inv = 1.f / fmaxf(msum, EPS);

  const int c = tid * 2;                       // 256 threads x 2 cols = 512
  const float* mp = m + (size_t)b * S_SRC * DIM + c;
  float ax = 0.f, ay = 0.f;
#pragma unroll 8
  for (int s = 0; s < S_SRC; ++s) {
    const float2 v = *(const float2*)(mp + (size_t)s * DIM);
    ax += w[s] * v.x;
    ay += w[s] * v.y;
  }
  ax *= inv; ay *= inv;
  *(float2*)(s_read + (size_t)b * DIM + c) = make_float2(ax, ay);

  red[tid] = ax * ax + ay * ay;
  __syncthreads();
  for (int off = 128; off > 0; off >>= 1) {
    if (tid < off) red[tid] += red[tid + off];
    __syncthreads();
  }
  if (tid == 0) s_sq[b] = red[0];
}

// ---------------------------------------------------------------------------
// Kernel 2a: tgt_w = L1-normalized tgt_mask rows (one block per (b,t) row)
// ---------------------------------------------------------------------------
__global__ __launch_bounds__(256) void k_tgt_weights(
    const int* __restrict__ tmask, float* __restrict__ tgt_w) {
  __shared__ float red[256];
  const int r = blockIdx.x, tid = threadIdx.x;
  const float v = (float)tmask[(size_t)r * TLEN + tid];
  red[tid] = v;
  __syncthreads();
  for (int off = 128; off > 0; off >>= 1) {
    if (tid < off) red[tid] += red[tid + off];
    __syncthreads();
  }
  tgt_w[(size_t)r * TLEN + tid] = v * (1.f / fmaxf(red[0], EPS));
}

// ---------------------------------------------------------------------------
// Kernel 2b: tgt_read[b] = tgt_w[b] (256x256) @ om[b] (256x512), f32 WMMA.
// One wave per 16x16 output tile; K loop in steps of 4 (V_WMMA_F32_16X16X4_F32).
// ---------------------------------------------------------------------------
__global__ __launch_bounds__(256) void k_tgt_readout_wmma(
    const float* __restrict__ tgt_w, const float* __restrict__ om,
    float* __restrict__ t_read) {
  const int tid = threadIdx.x;
  const int wid = blockIdx.x * 8 + (tid >> 5);     // 16384 waves total
  const int lane = tid & 31, lo = lane & 15, hi = lane >> 4;
  const int b = wid >> 9;                          // 512 tiles per batch
  const int rem = wid & 511;
  const int tt = rem >> 5;                         // 0..15 (t tile)
  const int td = rem & 31;                         // 0..31 (d tile)

  const float* W  = tgt_w + ((size_t)b * TLEN + tt * 16 + lo) * TLEN; // k contiguous
  const float* Bm = om + (size_t)b * TLEN * DIM + td * 16 + lo;       // k strided

  v8f c = {};
  for (int k0 = 0; k0 < TLEN; k0 += 4) {
    const int ka = k0 + 2 * hi;
    v2f a = *(const v2f*)(W + ka);
    v2f bb;
    bb.x = Bm[(size_t)ka * DIM];
    bb.y = Bm[(size_t)(ka + 1) * DIM];
    c = __builtin_amdgcn_wmma_f32_16x16x4_f32(false, a, false, bb, (short)0, c,
                                              false, false);
  }
  const size_t ob = ((size_t)b * TLEN + tt * 16) * DIM + td * 16 + lo;
#pragma unroll
  for (int i = 0; i < 8; ++i)
    t_read[ob + (size_t)(i + 8 * hi) * DIM] = c[i];
}

// ---------------------------------------------------------------------------
// Kernel 2c: t_sq[r] = sum_d t_read[r][d]^2
// ---------------------------------------------------------------------------
__global__ __launch_bounds__(256) void k_rowsumsq(
    const float* __restrict__ x, float* __restrict__ out) {
  __shared__ float red[256];
  const int r = blockIdx.x, tid = threadIdx.x;
  const float* p = x + (size_t)r * DIM;
  const float a = p[tid], b = p[tid + 256];
  red[tid] = a * a + b * b;
  __syncthreads();
  for (int off = 128; off > 0; off >>= 1) {
    if (tid < off) red[tid] += red[tid + off];
    __syncthreads();
  }
  if (tid == 0) out[r] = red[0];
}

// ---------------------------------------------------------------------------
// Kernel 3: cross GEMM [8192,512]x[512,4096] with fused distance/exp epilogue.
// Block owns 16 rows x ALL 4096 cols. The 16x512 f32 T-tile is staged into
// LDS by the Tensor Data Mover (one descriptor per block, TENSORcnt-tracked),
// with TDM padding of 2 dwords every 256 dwords -> row stride 516 floats
// (bank-conflict-free, 8B-aligned). S panel (8 MB) stays L2-resident.
// Row sums of sim accumulated in registers -> shfl reduce -> LDS -> ws.
// ---------------------------------------------------------------------------
__global__ __launch_bounds__(256) void k_cross_sim(
    const float* __restrict__ t_read, const float* __restrict__ s_read,
    const float* __restrict__ t_sq, const float* __restrict__ s_sq,
    float* __restrict__ out, float* __restrict__ rowsum) {
  __shared__ float A_lds[16 * A_STRIDE];
  __shared__ float ts_l[16];
  __shared__ float rs_l[16];
  const int tid = threadIdx.x;
  const int r0 = blockIdx.x * 16;

  if (tid == 0) {
    // --- Tensor DMA descriptor (CDNA5 ISA) ---
    const unsigned long long ga =
        (unsigned long long)(uintptr_t)(t_read + (size_t)r0 * DIM);
    const unsigned int lds = (unsigned int)(uintptr_t)(&A_lds[0]);
    v4u g0;
    g0.x = 1u;                                    // count=1, user mode, no gather
    g0.y = lds;                                   // lds_addr (bytes)
    g0.z = (unsigned int)(ga & 0xffffffffu);      // global_addr[31:0]
    g0.w = (unsigned int)((ga >> 32) & 0x1ffffffu) | (2u << 30); // addr[56:32]|type=2
    v8i g1;
    g1[0] = (2 << 16)        // data_size = 4 bytes
          | (1 << 20)        // pad_enable
          | (7 << 22)        // pad_interval: 256 dwords
          | (1 << 25);       // pad_amount: 2 dwords
    g1[1] = (DIM & 0xffff) << 16;                     // tensor_dim0[15:0]
    g1[2] = (DIM >> 16) | ((ROWS & 0xffff) << 16);    // dim0[31:16] | dim1[15:0]
    g1[3] = (ROWS >> 16) | (DIM << 16);               // dim1[31:16] | tile_dim0=512
    g1[4] = 16;                                       // tile_dim1=16, tile_dim2=0
    g1[5] = DIM;                                      // tensor_dim0_stride[31:0]
    g1[6] = 0;                                        // stride hi | dim1_stride lo
    g1[7] = 0;
    v4i gz4 = {0, 0, 0, 0};                           // groups 2/3 unused (2D)
    v8i gz8 = {0, 0, 0, 0, 0, 0, 0, 0};               // trailing group (unused)
    __builtin_amdgcn_tensor_load_to_lds(g0, g1, gz4, gz4, gz8, 0);
    __builtin_amdgcn_s_wait_tensorcnt(0);
  }
  if (tid < 16) { ts_l[tid] = t_sq[r0 + tid]; rs_l[tid] = 0.f; }
  __syncthreads();

  const int w = tid >> 5, lane = tid & 31, lo = lane & 15, hi = lane >> 4;
  v8f simsum = {};

  for (int ct = 0; ct < 32; ++ct) {          // 32 col-tiles of 128 (8 waves x 16)
    const int col = ct * 128 + w * 16 + lo;
    const float* S = s_read + (size_t)col * DIM;  // B = S^T: k contiguous per row
    __builtin_prefetch(S + (size_t)128 * DIM, 0, 1);  // next panel (L2-resident)
    v8f c = {};
#pragma unroll 4
    for (int k0 = 0; k0 < DIM; k0 += 4) {
      const int ka = k0 + 2 * hi;
      const int kofs = ka + ((ka >> 7) & 2);   // TDM mid-row pad at k=256
      v2f a  = *(const v2f*)(&A_lds[lo * A_STRIDE + kofs]);
      v2f bb = *(const v2f*)(S + ka);
      c = __builtin_amdgcn_wmma_f32_16x16x4_f32(false, a, false, bb, (short)0, c,
                                                false, false);
    }
    const float sc = s_sq[col];
#pragma unroll
    for (int i = 0; i < 8; ++i) {
      const int rl = i + 8 * hi;                       // WMMA C layout row
      float d2 = fmaxf(ts_l[rl] + sc - 2.f * c[i], EPS);
      float sim = expf(-sqrtf(d2));
      out[(size_t)(r0 + rl) * COLS + col] = sim;       // unnormalized for now
      simsum[i] += sim;
    }
  }

  // Reduce sim row-sums over the 16 lanes of each half-wave (cols of the tile)
#pragma unroll
  for (int off = 1; off < 16; off <<= 1) {
#pragma unroll
    for (int i = 0; i < 8; ++i) simsum[i] += __shfl_xor(simsum[i], off, 32);
  }
  if (lo == 0) {
#pragma unroll
    for (int i = 0; i < 8; ++i) atomicAdd(&rs_l[i + 8 * hi], simsum[i]);
  }
  __syncthreads();
  if (tid < 16) rowsum[r0 + tid] = rs_l[tid];
}

// ---------------------------------------------------------------------------
// Kernel 4: L1 normalize rows of the output (float4 per thread).
// ---------------------------------------------------------------------------
__global__ __launch_bounds__(256) void k_normalize(
    float* __restrict__ out, const float* __restrict__ rowsum) {
  const size_t i4 = ((size_t)blockIdx.x * blockDim.x + threadIdx.x) * 4;
  const int r = (int)(i4 >> 12);                 // 4096 cols per row
  const float inv = 1.f / fmaxf(rowsum[r], EPS);
  float4* p = (float4*)(out + i4);
  float4 v = *p;
  v.x *= inv; v.y *= inv; v.z *= inv; v.w *= inv;
  *p = v;
}

// ---------------------------------------------------------------------------
extern "C" void kernel_launch(void* const* d_in, const int* in_sizes, int n_in,
                              void* d_out, int out_size, void* d_ws, size_t ws_size,
                              hipStream_t stream) {
  const float* m     = (const float*)d_in[0];
  const float* om    = (const float*)d_in[1];
  const int*   smask = (const int*)d_in[2];
  const int*   tmask = (const int*)d_in[3];
  float* out = (float*)d_out;

  float* ws     = (float*)d_ws;
  float* s_read = ws;                                    // 4096*512
  float* t_read = s_read + (size_t)B_SRC * DIM;          // 8192*512
  float* s_sq   = t_read + (size_t)ROWS * DIM;           // 4096
  float* t_sq   = s_sq + B_SRC;                          // 8192
  float* tgt_w  = t_sq + ROWS;                           // 32*256*256
  float* rowsum = tgt_w + (size_t)B_TGT * TLEN * TLEN;   // 8192

  k_src_readout<<<B_SRC, 256, 0, stream>>>(m, smask, s_read, s_sq);
  k_tgt_weights<<<ROWS, 256, 0, stream>>>(tmask, tgt_w);
  k_tgt_readout_wmma<<<2048, 256, 0, stream>>>(tgt_w, om, t_read);
  k_rowsumsq<<<ROWS, 256, 0, stream>>>(t_read, t_sq);
  k_cross_sim<<<ROWS / 16, 256, 0, stream>>>(t_read, s_read, t_sq, s_sq, out, rowsum);
  k_normalize<<<(ROWS * (size_t)COLS) / (256 * 4), 256, 0, stream>>>(out, rowsum);
}